// XylemMixer_79628693668325
// MI455X (gfx1250) — compile-verified
//
#include <hip/hip_runtime.h>
#include <math.h>

typedef __attribute__((ext_vector_type(2))) float v2f;
typedef __attribute__((ext_vector_type(8))) float v8f;

#define B_ 2
#define T_ 2048
#define D_ 1024
#define H_ 16
#define K_ 64
#define P_ (3 * D_ + H_)      // 3088
#define SCALE_ 0.03125f       // 1/sqrt(1024)
#define WINDOW_ 1024          // exp(-0.1*1024) = e^-102 -> exactly 0 in fp32

// ---------------------------------------------------------------------------
// GEMM: C[M,N] = X[M,Kd] * W[N,Kd]^T   (all row-major, fp32, f32 WMMA 16x16x4)
// Block tile 128(M) x 64(N), K chunk 32, 8 wave32 waves.
// Wave (wm,wn): wm in {0,1} -> 64-row slab, wn in {0..3} -> 16-col slab.
// A frag (16x4 f32): lanes 0-15 rows M0-15 K={0,1}; lanes 16-31 K={2,3}.
// B frag (4x16 f32): lane%16 = N col; lanes 0-15 K={0,1}; lanes 16-31 K={2,3}.
// ---------------------------------------------------------------------------
#define GTM 128
#define GTN 64
#define GTK 32
#define LDSP 36   // padded LDS row stride (floats): 16B-aligned, kills bank conflicts

__global__ __launch_bounds__(256) void k_gemm(const float* __restrict__ X,
                                              const float* __restrict__ W,
                                              float* __restrict__ C,
                                              int M, int N, int Kd)
{
    __shared__ float Xs[GTM][LDSP];
    __shared__ float Ws[GTN][LDSP];

    const int tid   = threadIdx.x;
    const int lane  = tid & 31;
    const int wave  = tid >> 5;
    const int l16   = lane & 15;
    const int khalf = (lane < 16) ? 0 : 2;
    const int m0    = blockIdx.y * GTM;
    const int n0    = blockIdx.x * GTN;
    const int wm    = wave >> 2;
    const int wn    = wave & 3;

    v8f acc[4];
#pragma unroll
    for (int i = 0; i < 4; ++i)
#pragma unroll
        for (int r = 0; r < 8; ++r) acc[i][r] = 0.0f;

    for (int k0 = 0; k0 < Kd; k0 += GTK) {
        __syncthreads();
        // stage X tile: 128x32 floats = 1024 float4, coalesced
#pragma unroll
        for (int i = 0; i < 4; ++i) {
            int f4  = tid + i * 256;
            int row = f4 >> 3;
            int c4  = f4 & 7;
            *reinterpret_cast<float4*>(&Xs[row][c4 * 4]) =
                *reinterpret_cast<const float4*>(&X[(size_t)(m0 + row) * Kd + k0 + c4 * 4]);
        }
        // stage W tile: 64x32 floats = 512 float4, zero-pad past N
#pragma unroll
        for (int i = 0; i < 2; ++i) {
            int f4  = tid + i * 256;
            int row = f4 >> 3;
            int c4  = f4 & 7;
            float4 src = make_float4(0.f, 0.f, 0.f, 0.f);
            if (n0 + row < N)
                src = *reinterpret_cast<const float4*>(&W[(size_t)(n0 + row) * Kd + k0 + c4 * 4]);
            *reinterpret_cast<float4*>(&Ws[row][c4 * 4]) = src;
        }
        if (k0 + GTK < Kd)  // emits global_prefetch_b8 for the next X chunk
            __builtin_prefetch(&X[(size_t)(m0 + (tid & 127)) * Kd + k0 + GTK], 0, 0);
        __syncthreads();

#pragma unroll
        for (int k4 = 0; k4 < 8; ++k4) {
            const v2f b = *reinterpret_cast<const v2f*>(&Ws[wn * 16 + l16][k4 * 4 + khalf]);
#pragma unroll
            for (int i = 0; i < 4; ++i) {
                const v2f a =
                    *reinterpret_cast<const v2f*>(&Xs[wm * 64 + i * 16 + l16][k4 * 4 + khalf]);
                acc[i] = __builtin_amdgcn_wmma_f32_16x16x4_f32(
                    false, a, false, b, (short)0, acc[i], false, false);
            }
        }
    }

    // store: C/D layout -> VGPR r: lanes0-15 (M=r), lanes16-31 (M=r+8); N = lane%16
    const int rofs = (lane < 16) ? 0 : 8;
    const int col  = n0 + wn * 16 + l16;
    if (col < N) {
#pragma unroll
        for (int i = 0; i < 4; ++i)
#pragma unroll
            for (int r = 0; r < 8; ++r)
                C[(size_t)(m0 + wm * 64 + i * 16 + r + rofs) * N + col] = acc[i][r];
    }
}

// ---------------------------------------------------------------------------
// Depthwise causal conv (taps at t-6..t-3) + SiLU gate on v + decay logits
// ---------------------------------------------------------------------------
__global__ __launch_bounds__(256) void k_conv_gate(const float* __restrict__ proj,
                                                   const float* __restrict__ conv_w,
                                                   const float* __restrict__ conv_b,
                                                   const float* __restrict__ decay_param,
                                                   float* __restrict__ v_in,
                                                   float* __restrict__ ldb)
{
    const int bt = blockIdx.x;
    const int b  = bt / T_;
    const int t  = bt % T_;
    const size_t rowbase = (size_t)bt * P_;

    for (int d = threadIdx.x; d < D_; d += 256) {
        float kc = conv_b[d];
#pragma unroll
        for (int j = 0; j < 4; ++j) {
            int   tau = t - 6 + j;
            float kv  = (tau >= 0) ? proj[(size_t)(b * T_ + tau) * P_ + 2 * D_ + d] : 0.f;
            kc = fmaf(conv_w[d * 4 + j], kv, kc);
        }
        float sk = kc / (1.f + expf(-kc));          // silu(k_out)
        float v  = proj[rowbase + D_ + d] * sk;
        int   h  = d >> 6, kk = d & 63;
        v_in[((size_t)(b * H_ + h) * T_ + t) * K_ + kk] = v * SCALE_;
    }
    if (threadIdx.x < H_) {
        int   h = threadIdx.x;
        float w = proj[rowbase + 3 * D_ + h];
        float g = 1.f / (1.f + expf(-(decay_param[h] + w)));
        ldb[(size_t)(b * H_ + h) * T_ + t] = -8.f * (1.f - g) - 0.1f;
    }
}

// ---------------------------------------------------------------------------
// Inclusive cumsum of log-decay over T per (b,h): 256 threads x 8 elems + scan
// ---------------------------------------------------------------------------
__global__ __launch_bounds__(256) void k_cumsum(const float* __restrict__ ldb,
                                                float* __restrict__ csb)
{
    __shared__ float ssum[256];
    const int    tid  = threadIdx.x;
    const size_t base = (size_t)blockIdx.x * T_;

    float loc[8];
    float run = 0.f;
#pragma unroll
    for (int j = 0; j < 8; ++j) { run += ldb[base + tid * 8 + j]; loc[j] = run; }
    ssum[tid] = run;
    __syncthreads();
    for (int off = 1; off < 256; off <<= 1) {
        float v = (tid >= off) ? ssum[tid - off] : 0.f;
        __syncthreads();
        ssum[tid] += v;
        __syncthreads();
    }
    const float excl = ssum[tid] - run;
#pragma unroll
    for (int j = 0; j < 8; ++j) csb[base + tid * 8 + j] = loc[j] + excl;
}

// ---------------------------------------------------------------------------
// Windowed causal decay matmul: y[t,:] = sum_{s in [t-W, t]} exp(cs[t]-cs[s]) v[s,:]
// Block = 64 output rows x K=64 cols for one (b,h). A fragments are generated
// in-register from exp(); B fragments come from an LDS-staged v tile.
// ---------------------------------------------------------------------------
__global__ __launch_bounds__(256) void k_scan(const float* __restrict__ v_in,
                                              const float* __restrict__ csb,
                                              float* __restrict__ y)
{
    __shared__ float vs[64][64];
    __shared__ float cst[64];
    __shared__ float css[64];

    const int tid   = threadIdx.x;
    const int lane  = tid & 31;
    const int wave  = tid >> 5;
    const int l16   = lane & 15;
    const int khalf = (lane < 16) ? 0 : 2;
    const int bh    = blockIdx.y;
    const int t0    = blockIdx.x * 64;

    const float* cs = csb + (size_t)bh * T_;
    const float* vb = v_in + (size_t)bh * T_ * K_;

    const int mi = wave & 3;   // 16-row subtile
    const int nj = wave >> 2;  // 32-col half

    v8f acc[2];
#pragma unroll
    for (int i = 0; i < 2; ++i)
#pragma unroll
        for (int r = 0; r < 8; ++r) acc[i][r] = 0.0f;

    if (tid < 64) cst[tid] = cs[t0 + tid];
    const int tl = mi * 16 + l16;   // local output row (0..63)

    int s_lo = t0 - WINDOW_;
    if (s_lo < 0) s_lo = 0;

    for (int s0 = s_lo; s0 <= t0; s0 += 64) {
        __syncthreads();
#pragma unroll
        for (int i = 0; i < 4; ++i) {         // stage 64x64 v tile (coalesced float4)
            int f4  = tid + i * 256;
            int row = f4 >> 4;
            int c4  = f4 & 15;
            *reinterpret_cast<float4*>(&vs[row][c4 * 4]) =
                *reinterpret_cast<const float4*>(&vb[(size_t)(s0 + row) * K_ + c4 * 4]);
        }
        if (tid < 64) css[tid] = cs[s0 + tid];
        __syncthreads();

        const bool  diag = (s0 == t0);
        const float ct   = cst[tl];
#pragma unroll 4
        for (int k4 = 0; k4 < 16; ++k4) {
            const int ks = k4 * 4 + khalf;
            float ax = __expf(ct - css[ks]);
            float ay = __expf(ct - css[ks + 1]);
            if (diag) {                        // causal mask via data select (EXEC stays full)
                ax = (tl >= ks) ? ax : 0.f;
                ay = (tl >= ks + 1) ? ay : 0.f;
            }
            v2f a; a.x = ax; a.y = ay;
#pragma unroll
            for (int ni = 0; ni < 2; ++ni) {
                const int ncol = nj * 32 + ni * 16 + l16;
                v2f bf; bf.x = vs[ks][ncol]; bf.y = vs[ks + 1][ncol];
                acc[ni] = __builtin_amdgcn_wmma_f32_16x16x4_f32(
                    false, a, false, bf, (short)0, acc[ni], false, false);
            }
        }
    }

    const int rofs = (lane < 16) ? 0 : 8;
#pragma unroll
    for (int ni = 0; ni < 2; ++ni) {
        const int col = nj * 32 + ni * 16 + l16;
#pragma unroll
        for (int r = 0; r < 8; ++r)
            y[((size_t)bh * T_ + t0 + mi * 16 + r + rofs) * K_ + col] = acc[ni][r];
    }
}

// ---------------------------------------------------------------------------
// o = silu(r) * y, then GroupNorm over (K,T) per (b,h); writes (B,T,D) layout
// ---------------------------------------------------------------------------
__global__ __launch_bounds__(256) void k_gate_gn(const float* __restrict__ proj,
                                                 float* __restrict__ y,
                                                 const float* __restrict__ gn_w,
                                                 const float* __restrict__ gn_b,
                                                 float* __restrict__ onorm)
{
    __shared__ float s1[256], s2[256];
    __shared__ float mean_s, rstd_s;

    const int bh  = blockIdx.x;
    const int b   = bh / H_, h = bh % H_;
    const int tid = threadIdx.x;
    const size_t ybase = (size_t)bh * T_ * K_;

    float a1 = 0.f, a2 = 0.f;
    for (int it = 0; it < (T_ * K_) / 256; ++it) {
        int   flat = tid + it * 256;
        int   t = flat >> 6, kk = flat & 63;
        float r  = proj[(size_t)(b * T_ + t) * P_ + h * K_ + kk];
        float sr = r / (1.f + expf(-r));
        float o  = sr * y[ybase + flat];
        y[ybase + flat] = o;
        a1 += o; a2 += o * o;
    }
    s1[tid] = a1; s2[tid] = a2;
    __syncthreads();
    for (int off = 128; off > 0; off >>= 1) {
        if (tid < off) { s1[tid] += s1[tid + off]; s2[tid] += s2[tid + off]; }
        __syncthreads();
    }
    if (tid == 0) {
        float m   = s1[0] / (float)(T_ * K_);
        float var = s2[0] / (float)(T_ * K_) - m * m;
        mean_s = m;
        rstd_s = rsqrtf(var + 1e-5f);
    }
    __syncthreads();
    const float m = mean_s, rs = rstd_s;
    for (int it = 0; it < (T_ * K_) / 256; ++it) {
        int flat = tid + it * 256;
        int t = flat >> 6, kk = flat & 63;
        int d = h * K_ + kk;
        float o = y[ybase + flat];
        onorm[(size_t)(b * T_ + t) * D_ + d] = (o - m) * rs * gn_w[d] + gn_b[d];
    }
}

// ---------------------------------------------------------------------------
extern "C" void kernel_launch(void* const* d_in, const int* in_sizes, int n_in,
                              void* d_out, int out_size, void* d_ws, size_t ws_size,
                              hipStream_t stream)
{
    const float* x      = (const float*)d_in[0];
    const float* W_in   = (const float*)d_in[1];
    const float* conv_w = (const float*)d_in[2];
    const float* conv_b = (const float*)d_in[3];
    const float* decay  = (const float*)d_in[4];
    const float* gn_w   = (const float*)d_in[5];
    const float* gn_b   = (const float*)d_in[6];
    const float* W_out  = (const float*)d_in[7];
    float*       out    = (float*)d_out;

    // workspace layout (fp32):  proj | v_in | ld | cs | y | out_norm  (~101 MB)
    float* proj  = (float*)d_ws;
    float* v_in  = proj + (size_t)(B_ * T_) * P_;
    float* ldb   = v_in + (size_t)B_ * H_ * T_ * K_;
    float* csb   = ldb + (size_t)B_ * H_ * T_;
    float* yb    = csb + (size_t)B_ * H_ * T_;
    float* onorm = yb + (size_t)B_ * H_ * T_ * K_;

    {   // GEMM1: proj = x @ W_in^T   (4096 x 3088 x 1024)
        dim3 grid((P_ + GTN - 1) / GTN, (B_ * T_) / GTM);
        k_gemm<<<grid, 256, 0, stream>>>(x, W_in, proj, B_ * T_, P_, D_);
    }
    k_conv_gate<<<B_ * T_, 256, 0, stream>>>(proj, conv_w, conv_b, decay, v_in, ldb);
    k_cumsum<<<B_ * H_, 256, 0, stream>>>(ldb, csb);
    {   // windowed causal decay matmul
        dim3 grid(T_ / 64, B_ * H_);
        k_scan<<<grid, 256, 0, stream>>>(v_in, csb, yb);
    }
    k_gate_gn<<<B_ * H_, 256, 0, stream>>>(proj, yb, gn_w, gn_b, onorm);
    {   // GEMM3: out = onorm @ W_out^T   (4096 x 1024 x 1024)
        dim3 grid(D_ / GTN, (B_ * T_) / GTM);
        k_gemm<<<grid, 256, 0, stream>>>(onorm, W_out, out, B_ * T_, D_, D_);
    }
}